// HorizontalComparisonModel_84593675862345
// MI455X (gfx1250) — compile-verified
//
#include <hip/hip_runtime.h>
#include <hip/hip_bf16.h>

#ifndef __has_builtin
#define __has_builtin(x) 0
#endif

// Problem geometry (from reference): B=32, C=64, D=300, L=128
#define BQ 32
#define CQ 64
#define DQ 300
#define LQ 128

#define THREADS 128           // one thread per l-column
#define CHUNK 20              // d-rows per staged chunk (300 = 15 * 20)
#define NCHUNK (DQ / CHUNK)   // 15
#define CHUNK_FLOATS (CHUNK * LQ)          // 2560 floats = 10240 bytes per input
#define GRAN_PER_THREAD (CHUNK_FLOATS * 4 / 16 / THREADS)  // 5 x b128 per thread per input

typedef int v4i __attribute__((ext_vector_type(4)));
typedef __attribute__((address_space(1))) v4i* g_v4i_p;   // global (device) int4*
typedef __attribute__((address_space(3))) void* gas3_t;   // LDS void*
typedef __attribute__((address_space(3))) v4i* l_v4i_p;   // LDS int4*

// ---- CDNA5 async datamover: global -> LDS, tracked by ASYNCcnt ----
__device__ __forceinline__ void async_copy_b128(const float* gsrc, float* ldst) {
#if __has_builtin(__builtin_amdgcn_global_load_async_to_lds_b128)
    __builtin_amdgcn_global_load_async_to_lds_b128(
        (g_v4i_p)(unsigned long long)gsrc,   // global AS1 address == flat address
        (l_v4i_p)(gas3_t)ldst,               // addrspacecast generic->AS3, then bitcast
        0, 0);
#else
    unsigned lds_off = (unsigned)(unsigned long long)(gas3_t)ldst;
    asm volatile("global_load_async_to_lds_b128 %0, %1, off"
                 :: "v"(lds_off), "v"(gsrc)
                 : "memory");
#endif
}

template <int N>
__device__ __forceinline__ void wait_asynccnt_le() {
#if __has_builtin(__builtin_amdgcn_s_wait_asynccnt)
    __builtin_amdgcn_s_wait_asynccnt(N);
#else
    asm volatile("s_wait_asynccnt %0" :: "i"(N) : "memory");
#endif
}

__device__ __forceinline__ void issue_chunk(const float* gx, const float* gy,
                                            float* lx, float* ly, int t) {
#pragma unroll
    for (int i = 0; i < GRAN_PER_THREAD; ++i) {
        const int f = (i * THREADS + t) * 4;   // float index of this 16B granule
        async_copy_b128(gx + f, lx + f);
        async_copy_b128(gy + f, ly + f);
    }
}

__global__ __launch_bounds__(THREADS)
void HorizontalComparisonModel_84593675862345_kernel(const float* __restrict__ x,
                                                     const float* __restrict__ y,
                                                     float* __restrict__ out) {
    const int bc = blockIdx.x;        // 0 .. B*C-1, selects a contiguous [D, L] slab
    const int t  = threadIdx.x;       // l-column owned by this thread

    const float* gx = x + (size_t)bc * (DQ * LQ);
    const float* gy = y + (size_t)bc * (DQ * LQ);

    // Double-buffered staging: 2 bufs * 2 inputs * 10240B = 40 KB LDS / block
    __shared__ __align__(16) float bufx[2][CHUNK_FLOATS];
    __shared__ __align__(16) float bufy[2][CHUNK_FLOATS];

    // Prologue: prefetch chunk 0 into buffer 0
    issue_chunk(gx, gy, bufx[0], bufy[0], t);

    float sxx = 0.f, syy = 0.f, sxy = 0.f, sdd = 0.f, sad = 0.f;

    for (int k = 0; k < NCHUNK; ++k) {
        const int cur = k & 1;

        if (k + 1 < NCHUNK) {
            // Prefetch next chunk into the other buffer, then wait until only
            // those 2*GRAN_PER_THREAD fresh ops remain outstanding -> chunk k landed
            // (async loads complete in order).
            issue_chunk(gx + (size_t)(k + 1) * CHUNK_FLOATS,
                        gy + (size_t)(k + 1) * CHUNK_FLOATS,
                        bufx[cur ^ 1], bufy[cur ^ 1], t);
            wait_asynccnt_le<2 * GRAN_PER_THREAD>();
        } else {
            wait_asynccnt_le<0>();
        }
        __syncthreads();   // chunk k visible to all 4 waves

        // Row-wise partial reductions; lanes 0..31 hit 32 distinct LDS banks.
#pragma unroll
        for (int dd = 0; dd < CHUNK; ++dd) {
            const float xv = bufx[cur][dd * LQ + t];
            const float yv = bufy[cur][dd * LQ + t];
            sxx = fmaf(xv, xv, sxx);
            syy = fmaf(yv, yv, syy);
            sxy = fmaf(xv, yv, sxy);
            const float dv = xv - yv;
            sdd = fmaf(dv, dv, sdd);
            sad += fabsf(dv);
        }
        __syncthreads();   // all waves done reading buf[cur] before it is re-issued
    }

    // Epilogue (matches reference semantics exactly)
    const float nx  = fmaxf(sqrtf(sxx), 1e-12f);          // COS_EPS
    const float ny  = fmaxf(sqrtf(syy), 1e-12f);
    const float cosv = sxy / (nx * ny);
    const float sq  = sdd + ((sdd < 1e-8f) ? 1e-8f : 0.f); // revise_zero_data
    const float l2  = sqrtf(sq);

    const size_t o = ((size_t)bc * LQ + t) * 3;            // [B, C*L, 3] flat
    out[o + 0] = cosv;
    out[o + 1] = l2;
    out[o + 2] = sad;
}

extern "C" void kernel_launch(void* const* d_in, const int* in_sizes, int n_in,
                              void* d_out, int out_size, void* d_ws, size_t ws_size,
                              hipStream_t stream) {
    const float* x = (const float*)d_in[0];   // input_data1, fp32 [B,C,D,L]
    const float* y = (const float*)d_in[1];   // input_data2, fp32 [B,C,D,L]
    float* out = (float*)d_out;               // fp32 [B, C*L, 3]

    dim3 grid(BQ * CQ);     // 2048 blocks, one per (b, c) slab
    dim3 block(THREADS);    // 128 threads = 4 wave32
    HorizontalComparisonModel_84593675862345_kernel<<<grid, block, 0, stream>>>(x, y, out);
}